// GCNEncoder_10101763080453
// MI455X (gfx1250) — compile-verified
//
#include <hip/hip_runtime.h>

typedef __attribute__((ext_vector_type(2))) float v2f;
typedef __attribute__((ext_vector_type(8))) float v8f;

#define D 128
#define BN_EPS 1e-5f

// ---------------------------------------------------------------------------
// Degree / normalization
// ---------------------------------------------------------------------------
__global__ void k_deg_init(float* __restrict__ deg, int n) {
  int i = blockIdx.x * blockDim.x + threadIdx.x;
  if (i < n) deg[i] = 1.0f;  // self-loop contributes 1 to every node
}

__global__ void k_deg_count(const int* __restrict__ dst, float* __restrict__ deg, int e) {
  int i = blockIdx.x * blockDim.x + threadIdx.x;
  if (i < e) atomicAdd(&deg[dst[i]], 1.0f);
}

__global__ void k_dinv(float* __restrict__ deg, int n) {
  int i = blockIdx.x * blockDim.x + threadIdx.x;
  if (i < n) {
    float d = deg[i];
    deg[i] = (d > 0.0f) ? rsqrtf(d) : 0.0f;  // overwrite in place with dinv
  }
}

// ---------------------------------------------------------------------------
// GEMM: C[n x 128] = A[n x 128] @ B[128 x 128] (+ bias), fp32 WMMA 16x16x4
// blockDim = 256 (8 waves); wave w owns column-tile w (8 tiles of 16 cover D).
// A-frag (16x4, ISA layout): lane -> M = lane&15, VGPR0/1 = K pair (lane>>4)*2.
// B-frag (4x16): lane -> N = lane&15, VGPR0/1 = K pair (lane>>4)*2.
// C/D (16x16): vgpr r -> row r + (lane>>4)*8, col = lane&15.
// ---------------------------------------------------------------------------
__global__ void k_gemm_wmma(const float* __restrict__ A, const float* __restrict__ B,
                            const float* __restrict__ bias, float* __restrict__ C,
                            int nrows) {
  const int lane = threadIdx.x & 31;
  const int wave = threadIdx.x >> 5;
  const int rowbase = blockIdx.x << 4;
  const int colbase = wave << 4;
  if (rowbase >= nrows) return;

  const int m = lane & 15;
  const int kh = (lane >> 4) << 1;  // 0 or 2

  const float* __restrict__ arow = A + (size_t)(rowbase + m) * D;

  v8f c = {};
#pragma unroll
  for (int k0 = 0; k0 < D; k0 += 4) {
    const int kk = k0 + kh;
    v2f a;
    a.x = arow[kk];
    a.y = arow[kk + 1];
    v2f b;
    b.x = B[(size_t)kk * D + colbase + m];
    b.y = B[(size_t)(kk + 1) * D + colbase + m];
    c = __builtin_amdgcn_wmma_f32_16x16x4_f32(false, a, false, b, (short)0, c,
                                              false, false);
  }

  const float bv = bias ? bias[colbase + m] : 0.0f;
  const int row0 = rowbase + ((lane >> 4) << 3);
#pragma unroll
  for (int r = 0; r < 8; ++r) {
    C[(size_t)(row0 + r) * D + colbase + m] = c[r] + bv;
  }
}

// ---------------------------------------------------------------------------
// agg = h * dinv^2 (self-loop term) + bias   (initializes the scatter target)
// ---------------------------------------------------------------------------
__global__ void k_agg_init(const float* __restrict__ h, const float* __restrict__ dinv,
                           const float* __restrict__ bias, float* __restrict__ agg,
                           int total) {
  int idx = blockIdx.x * blockDim.x + threadIdx.x;
  if (idx >= total) return;
  int d = idx & (D - 1);
  int i = idx >> 7;
  float di = dinv[i];
  agg[idx] = h[idx] * di * di + bias[d];
}

// ---------------------------------------------------------------------------
// Edge scatter: one wave per edge; each lane handles 4 contiguous floats.
// agg[dst] += h[src] * dinv[src]*dinv[dst]   (float atomics, land in L2)
// ---------------------------------------------------------------------------
__global__ void k_edge_scatter(const float* __restrict__ h, const int* __restrict__ src,
                               const int* __restrict__ dst, const float* __restrict__ dinv,
                               float* __restrict__ agg, int e) {
  int wv = (blockIdx.x * blockDim.x + threadIdx.x) >> 5;
  int lane = threadIdx.x & 31;
  if (wv >= e) return;
  int s = src[wv];
  int t = dst[wv];
  float w = dinv[s] * dinv[t];
  const float4* __restrict__ hs = (const float4*)(h + (size_t)s * D);
  float4 v = hs[lane];
  float* __restrict__ out = agg + (size_t)t * D + lane * 4;
  atomicAdd(out + 0, v.x * w);
  atomicAdd(out + 1, v.y * w);
  atomicAdd(out + 2, v.z * w);
  atomicAdd(out + 3, v.w * w);
}

// ---------------------------------------------------------------------------
// BatchNorm statistics: sums[0:128] = column sums, sums[128:256] = sum of sq.
// blockDim = 128 (one thread per column), each block accumulates 256 rows
// locally then does one atomicAdd per column.
// ---------------------------------------------------------------------------
__global__ void k_zero_sums(float* __restrict__ sums) {
  sums[threadIdx.x] = 0.0f;  // launched with 256 threads
}

__global__ void k_colsum(const float* __restrict__ h, float* __restrict__ sums, int n) {
  int d = threadIdx.x;  // 0..127
  int r0 = blockIdx.x * 256;
  int r1 = r0 + 256;
  if (r1 > n) r1 = n;
  float s = 0.0f, s2 = 0.0f;
  for (int r = r0; r < r1; ++r) {
    float v = h[(size_t)r * D + d];
    s += v;
    s2 += v * v;
  }
  atomicAdd(&sums[d], s);
  atomicAdd(&sums[D + d], s2);
}

// ---------------------------------------------------------------------------
// Fused BatchNorm (biased variance, matching jnp.var) + ReLU
// ---------------------------------------------------------------------------
__global__ void k_bn_relu(const float* __restrict__ h, const float* __restrict__ sums,
                          const float* __restrict__ g, const float* __restrict__ be,
                          float* __restrict__ out, int n, int total) {
  int idx = blockIdx.x * blockDim.x + threadIdx.x;
  if (idx >= total) return;
  int d = idx & (D - 1);
  float inv_n = 1.0f / (float)n;
  float mu = sums[d] * inv_n;
  float var = sums[D + d] * inv_n - mu * mu;
  float sc = g[d] * rsqrtf(var + BN_EPS);
  float v = (h[idx] - mu) * sc + be[d];
  out[idx] = v > 0.0f ? v : 0.0f;
}

// ---------------------------------------------------------------------------
// Host-side orchestration
// ---------------------------------------------------------------------------
static inline size_t align256(size_t x) { return (x + 255) & ~(size_t)255; }

extern "C" void kernel_launch(void* const* d_in, const int* in_sizes, int n_in,
                              void* d_out, int out_size, void* d_ws, size_t ws_size,
                              hipStream_t stream) {
  const float* x  = (const float*)d_in[0];
  const int*  ei  = (const int*)d_in[1];   // (2, E) flat: [0..E) = src, [E..2E) = dst
  const float* W1 = (const float*)d_in[2];
  const float* b1 = (const float*)d_in[3];
  const float* g1 = (const float*)d_in[4];
  const float* be1= (const float*)d_in[5];
  const float* W2 = (const float*)d_in[6];
  const float* b2 = (const float*)d_in[7];
  const float* g2 = (const float*)d_in[8];
  const float* be2= (const float*)d_in[9];
  const float* Wf = (const float*)d_in[10];
  const float* bf = (const float*)d_in[11];
  float* out = (float*)d_out;

  const int N = in_sizes[0] / D;
  const int E = in_sizes[1] / 2;
  const int total = N * D;
  const int* src = ei;
  const int* dst = ei + E;

  // workspace layout
  char* ws = (char*)d_ws;
  float* dinv = (float*)ws;                                   // N floats
  size_t off = align256((size_t)N * sizeof(float));
  float* sums = (float*)(ws + off);                           // 256 floats
  off += align256(2 * D * sizeof(float));
  float* bufG = (float*)(ws + off);                           // N*D floats
  off += (size_t)total * sizeof(float);
  float* bufA = (float*)(ws + off);                           // N*D floats

  const int TB = 256;
  const int gN   = (N + TB - 1) / TB;
  const int gE   = (E + TB - 1) / TB;
  const int gEw  = (E * 32 + TB - 1) / TB;      // one wave per edge
  const int gTot = (total + TB - 1) / TB;
  const int gGemm = (N + 15) / 16;              // 16-row tiles, 8 col-tiles/block
  const int gCol  = (N + 255) / 256;

  // --- normalization ---
  k_deg_init<<<gN, TB, 0, stream>>>(dinv, N);
  k_deg_count<<<gE, TB, 0, stream>>>(dst, dinv, E);
  k_dinv<<<gN, TB, 0, stream>>>(dinv, N);

  // --- layer 1 ---
  k_gemm_wmma<<<gGemm, TB, 0, stream>>>(x, W1, nullptr, bufG, N);
  k_agg_init<<<gTot, TB, 0, stream>>>(bufG, dinv, b1, bufA, total);
  k_edge_scatter<<<gEw, TB, 0, stream>>>(bufG, src, dst, dinv, bufA, E);
  k_zero_sums<<<1, 256, 0, stream>>>(sums);
  k_colsum<<<gCol, D, 0, stream>>>(bufA, sums, N);
  k_bn_relu<<<gTot, TB, 0, stream>>>(bufA, sums, g1, be1, bufG, N, total);  // h1 -> bufG

  // --- layer 2 ---
  k_gemm_wmma<<<gGemm, TB, 0, stream>>>(bufG, W2, nullptr, bufA, N);
  k_agg_init<<<gTot, TB, 0, stream>>>(bufA, dinv, b2, bufG, total);
  k_edge_scatter<<<gEw, TB, 0, stream>>>(bufA, src, dst, dinv, bufG, E);
  k_zero_sums<<<1, 256, 0, stream>>>(sums);
  k_colsum<<<gCol, D, 0, stream>>>(bufG, sums, N);
  k_bn_relu<<<gTot, TB, 0, stream>>>(bufG, sums, g2, be2, bufA, N, total);  // h2 -> bufA

  // --- final projection (bias fused into WMMA epilogue) ---
  k_gemm_wmma<<<gGemm, TB, 0, stream>>>(bufA, Wf, bf, out, N);
}